// Net_22582938042866
// MI455X (gfx1250) — compile-verified
//
#include <hip/hip_runtime.h>
#include <math.h>

#define N_FEAT 512
#define HID    16
#define NCLS   7

typedef __attribute__((ext_vector_type(2))) float v2f;
typedef __attribute__((ext_vector_type(8))) float v8f;

// ---------------------------------------------------------------- degree ----
__global__ void k_deg_init(float* __restrict__ deg, int n) {
    int i = blockIdx.x * blockDim.x + threadIdx.x;
    if (i < n) deg[i] = 1.0f;               // self-loop contributes 1
}

__global__ void k_deg_edges(const int* __restrict__ col, float* __restrict__ deg, int E) {
    int e = blockIdx.x * blockDim.x + threadIdx.x;
    if (e < E) atomicAdd(&deg[col[e]], 1.0f);
}

__global__ void k_dinv(float* __restrict__ deg, int n) {
    int i = blockIdx.x * blockDim.x + threadIdx.x;
    if (i < n) deg[i] = rsqrtf(deg[i]);     // deg >= 1 always (self-loops)
}

// -------------------------------------------- layer 1 GEMM: h = x @ W1 -----
// One wave32 per 16-row tile; K marched in steps of 4 with V_WMMA_F32_16X16X4_F32.
// A frag (16x4 f32): lane = M (0..15), lane-half selects K-pair; a[v] = x[M][k + 2*kh + v]
// B frag (4x16 f32): lane = N, mirrored;  b[v] = W1[k + 2*kh + v][N]
// C/D  (16x16 f32): 8 VGPRs; M = v + 8*(lane>=16), N = lane & 15
__global__ __launch_bounds__(256) void k_gemm1(
    const float* __restrict__ x, const float* __restrict__ W1,
    const float* __restrict__ dinv,
    float* __restrict__ h, float* __restrict__ agg1, int nTiles)
{
    __shared__ float sW[N_FEAT * HID];          // 32 KB: W1 staged once per block
    for (int i = threadIdx.x; i < N_FEAT * HID; i += blockDim.x) sW[i] = W1[i];
    __syncthreads();

    int wave = threadIdx.x >> 5;
    int lane = threadIdx.x & 31;
    int tile = blockIdx.x * 8 + wave;
    if (tile >= nTiles) return;                 // wave-uniform: EXEC stays all-1s below

    int m  = lane & 15;
    int kh = lane >> 4;                         // 0 or 1 -> K offset 0 / 2
    const float* xr = x + (size_t)(tile * 16 + m) * N_FEAT + 2 * kh;

    v8f acc = {};
#pragma unroll 4
    for (int k = 0; k < N_FEAT; k += 4) {
        v2f a, b;
        a.x = xr[k];
        a.y = xr[k + 1];
        int kk = k + 2 * kh;
        b.x = sW[kk * HID + m];
        b.y = sW[(kk + 1) * HID + m];
        acc = __builtin_amdgcn_wmma_f32_16x16x4_f32(
            /*neg_a=*/false, a, /*neg_b=*/false, b,
            /*c_mod=*/(short)0, acc, /*reuse_a=*/false, /*reuse_b=*/false);
    }

    // Store tile; also fold the self-loop term of the aggregation: agg = h * dinv^2
#pragma unroll
    for (int v = 0; v < 8; ++v) {
        int node  = tile * 16 + v + 8 * kh;
        float val = acc[v];
        float di  = dinv[node];
        h[(size_t)node * HID + m]    = val;
        agg1[(size_t)node * HID + m] = val * di * di;
    }
}

// ------------------------------------------ layer 1 edge scatter-add -------
__global__ void k_edge1(const int* __restrict__ eidx, const float* __restrict__ dinv,
                        const float* __restrict__ h, float* __restrict__ agg1, int E)
{
    int t = blockIdx.x * blockDim.x + threadIdx.x;
    int e = t >> 4, f = t & 15;
    if (e >= E) return;
    int r = eidx[e];
    int c = eidx[E + e];
    float w = dinv[r] * dinv[c];
    atomicAdd(&agg1[(size_t)c * HID + f], h[(size_t)r * HID + f] * w);
}

// ------------- relu(agg1+b1), layer-2 transform, self-loop init of d_out ---
__global__ void k_l2(const float* __restrict__ agg1, const float* __restrict__ b1,
                     const float* __restrict__ W2, const float* __restrict__ dinv,
                     float* __restrict__ o, float* __restrict__ outInit, int n)
{
    __shared__ float sW2[HID * NCLS];
    __shared__ float sb1[HID];
    if (threadIdx.x < HID * NCLS) sW2[threadIdx.x] = W2[threadIdx.x];
    if (threadIdx.x < HID)        sb1[threadIdx.x] = b1[threadIdx.x];
    __syncthreads();

    int i = blockIdx.x * blockDim.x + threadIdx.x;
    if (i >= n) return;

    float hv[HID];
#pragma unroll
    for (int f = 0; f < HID; ++f) {
        float v = agg1[(size_t)i * HID + f] + sb1[f];
        hv[f] = v > 0.0f ? v : 0.0f;
    }
    float di = dinv[i];
    float d2 = di * di;
#pragma unroll
    for (int j = 0; j < NCLS; ++j) {
        float s = 0.0f;
#pragma unroll
        for (int f = 0; f < HID; ++f) s = fmaf(hv[f], sW2[f * NCLS + j], s);
        o[(size_t)i * NCLS + j]       = s;
        outInit[(size_t)i * NCLS + j] = s * d2;   // self-loop term
    }
}

// ------------------------------------------ layer 2 edge scatter-add -------
__global__ void k_edge2(const int* __restrict__ eidx, const float* __restrict__ dinv,
                        const float* __restrict__ o, float* __restrict__ out, int E)
{
    int t = blockIdx.x * blockDim.x + threadIdx.x;
    int e = t >> 3, j = t & 7;
    if (e >= E || j >= NCLS) return;
    int r = eidx[e];
    int c = eidx[E + e];
    float w = dinv[r] * dinv[c];
    atomicAdd(&out[(size_t)c * NCLS + j], o[(size_t)r * NCLS + j] * w);
}

// ------------------------------------------ +b2 and in-place log_softmax ---
__global__ void k_lsm(float* __restrict__ out, const float* __restrict__ b2, int n)
{
    int i = blockIdx.x * blockDim.x + threadIdx.x;
    if (i >= n) return;
    float v[NCLS];
    float mx = -1e30f;
#pragma unroll
    for (int j = 0; j < NCLS; ++j) {
        v[j] = out[(size_t)i * NCLS + j] + b2[j];
        mx = fmaxf(mx, v[j]);
    }
    float s = 0.0f;
#pragma unroll
    for (int j = 0; j < NCLS; ++j) s += __expf(v[j] - mx);
    float l = __logf(s);
#pragma unroll
    for (int j = 0; j < NCLS; ++j) out[(size_t)i * NCLS + j] = v[j] - mx - l;
}

// ---------------------------------------------------------------------------
extern "C" void kernel_launch(void* const* d_in, const int* in_sizes, int n_in,
                              void* d_out, int out_size, void* d_ws, size_t ws_size,
                              hipStream_t stream)
{
    const float* x    = (const float*)d_in[0];
    const int*   eidx = (const int*)  d_in[1];   // [2, E] flat: row then col
    const float* W1   = (const float*)d_in[2];
    const float* b1   = (const float*)d_in[3];
    const float* W2   = (const float*)d_in[4];
    const float* b2   = (const float*)d_in[5];
    float* out = (float*)d_out;

    const int N = in_sizes[0] / N_FEAT;
    const int E = in_sizes[1] / 2;

    float* ws   = (float*)d_ws;
    float* dinv = ws;                       // N floats (deg -> dinv in place)
    float* h    = ws + (size_t)N;           // N*16
    float* agg1 = ws + (size_t)N * 17;      // N*16
    float* obuf = ws + (size_t)N * 33;      // N*7

    const int B = 256;
    k_deg_init <<<(N + B - 1) / B, B, 0, stream>>>(dinv, N);
    k_deg_edges<<<(E + B - 1) / B, B, 0, stream>>>(eidx + E, dinv, E);
    k_dinv     <<<(N + B - 1) / B, B, 0, stream>>>(dinv, N);

    const int nTiles = (N + 15) / 16;       // 6250 for N=100000
    k_gemm1<<<(nTiles + 7) / 8, 256, 0, stream>>>(x, W1, dinv, h, agg1, nTiles);

    const int t1 = E * 16;                  // 51.2M threads: (edge, feature)
    k_edge1<<<(t1 + B - 1) / B, B, 0, stream>>>(eidx, dinv, h, agg1, E);

    k_l2<<<(N + B - 1) / B, B, 0, stream>>>(agg1, b1, W2, dinv, obuf, out, N);

    const int t2 = E * 8;                   // (edge, class) padded to 8
    k_edge2<<<(t2 + B - 1) / B, B, 0, stream>>>(eidx, dinv, obuf, out, E);

    k_lsm<<<(N + B - 1) / B, B, 0, stream>>>(out, b2, N);
}